// AdditiveAttention_33930241639015
// MI455X (gfx1250) — compile-verified
//
#include <hip/hip_runtime.h>
#include <math.h>

// Problem constants (match reference)
#define BB   8
#define MM   512
#define NN   512
#define HH   64
#define QSS  256
#define VSS  256
#define NEG_INF_F (-1e20f)

typedef float v2f __attribute__((ext_vector_type(2)));
typedef float v8f __attribute__((ext_vector_type(8)));

// ---------------------------------------------------------------------------
// Generic f32 WMMA GEMM: C[M,N] = A[M,K] @ B[K,N], row-major, one wave per
// 16x16 C tile, K-loop in steps of 4 using V_WMMA_F32_16X16X4_F32.
// grid = (Mtiles, Ntiles, batch), block = 32 (one wave, EXEC all ones).
//
// ISA 7.12.2 fragment layouts (wave32):
//   A 16x4 : lanes 0-15 -> M=lane, VGPR{0,1}=K{0,1}; lanes 16-31 -> K{2,3}
//   B 4x16 : VGPR0: lanes0-15 K=0 / lanes16-31 K=2; VGPR1: K=1 / K=3 (N=lane%16)
//   C 16x16: VGPR v: lanes0-15 -> M=v,  N=lane; lanes16-31 -> M=v+8, N=lane-16
// ---------------------------------------------------------------------------
__global__ __launch_bounds__(32)
void gemm_wmma_f32_16x16x4(const float* __restrict__ A,
                           const float* __restrict__ B,
                           float* __restrict__ C,
                           int lda, int ldb, int ldc, int K,
                           long sA, long sB, long sC) {
  const int lane = threadIdx.x;        // 0..31
  const int half = lane >> 4;          // 0 or 1
  const int l    = lane & 15;          // 0..15
  const int m0   = blockIdx.x * 16;
  const int n0   = blockIdx.y * 16;

  const float* Ab = A + (size_t)blockIdx.z * (size_t)sA;
  const float* Bb = B + (size_t)blockIdx.z * (size_t)sB;
  float*       Cb = C + (size_t)blockIdx.z * (size_t)sC;

  v8f acc = {0.f, 0.f, 0.f, 0.f, 0.f, 0.f, 0.f, 0.f};

  const float* arow = Ab + (size_t)(m0 + l) * (size_t)lda;   // row m0+l of A
  for (int k = 0; k < K; k += 4) {
    v2f a, b;
    // A fragment: this lane's row, columns k + 2*half + {0,1}
    const float* ap = arow + k + 2 * half;
    a.x = ap[0];
    a.y = ap[1];
    // B fragment: column n0+l, rows k + 2*half + {0,1}
    const float* bp = Bb + (size_t)(k + 2 * half) * (size_t)ldb + (n0 + l);
    b.x = bp[0];
    b.y = bp[ldb];
    // 8-arg form: (neg_a, A, neg_b, B, c_mod, C, reuse_a, reuse_b)
    acc = __builtin_amdgcn_wmma_f32_16x16x4_f32(false, a, false, b,
                                                (short)0, acc, false, false);
  }

  const int cn = n0 + l;
  const int cm = m0 + half * 8;
#pragma unroll
  for (int v = 0; v < 8; ++v) {
    Cb[(size_t)(cm + v) * (size_t)ldc + cn] = acc[v];
  }
}

// ---------------------------------------------------------------------------
// Fused additive-attention scores + mask + softmax.
// One wave per (b, m) query row; block = 256 threads = 8 waves = 8 rows.
// Each lane owns 16 of the 512 key positions (n = lane + 32*j), keeps the
// scores in registers, does the masked softmax with wave32 shuffles, and
// writes the attention row to the workspace.
//   scores[n] = sum_h tanh(Qt[b,m,h] + Kt[b,n,h]) * wv[h]
// ---------------------------------------------------------------------------
__global__ __launch_bounds__(256)
void additive_scores_softmax(const float* __restrict__ Qt,
                             const float* __restrict__ Kt,
                             const float* __restrict__ wv,
                             const int*   __restrict__ valid_lens,
                             float* __restrict__ attn) {
  __shared__ float s_wv[HH];
  __shared__ float s_q[8][HH];

  const int tid  = threadIdx.x;
  const int wave = tid >> 5;
  const int lane = tid & 31;

  if (tid < HH) s_wv[tid] = wv[tid];

  const int row = blockIdx.x * 8 + wave;   // 0..B*M-1
  const int b   = row >> 9;                // / 512
  const float* qrow = Qt + (size_t)row * HH;
  s_q[wave][lane]      = qrow[lane];
  s_q[wave][lane + 32] = qrow[lane + 32];
  __syncthreads();

  const int vlen = valid_lens[b];
  const float* ktb = Kt + (size_t)b * NN * HH;

  float sc[16];
#pragma unroll
  for (int j = 0; j < 16; ++j) {
    const int n = lane + 32 * j;
    const float* krow = ktb + (size_t)n * HH;
    float s = 0.f;
#pragma unroll 8
    for (int h = 0; h < HH; ++h) {
      s += tanhf(s_q[wave][h] + krow[h]) * s_wv[h];
    }
    sc[j] = (n < vlen) ? s : NEG_INF_F;
  }

  // masked softmax over the 512 positions this wave owns
  float mx = sc[0];
#pragma unroll
  for (int j = 1; j < 16; ++j) mx = fmaxf(mx, sc[j]);
#pragma unroll
  for (int o = 16; o > 0; o >>= 1) mx = fmaxf(mx, __shfl_xor(mx, o, 32));

  float sum = 0.f;
#pragma unroll
  for (int j = 0; j < 16; ++j) { sc[j] = expf(sc[j] - mx); sum += sc[j]; }
#pragma unroll
  for (int o = 16; o > 0; o >>= 1) sum += __shfl_xor(sum, o, 32);
  const float inv = 1.0f / sum;

  float* arow = attn + (size_t)row * NN;
#pragma unroll
  for (int j = 0; j < 16; ++j) arow[lane + 32 * j] = sc[j] * inv;
}

// ---------------------------------------------------------------------------
// Launch: Qt = Q@Wq ; Kt = K@Wk ; attn = softmax(mask(scores)) ; out = attn@V
// ws layout: Qt (B*M*H) | Kt (B*N*H) | attn (B*M*N)   -> 10 MB of f32
// ---------------------------------------------------------------------------
extern "C" void kernel_launch(void* const* d_in, const int* in_sizes, int n_in,
                              void* d_out, int out_size, void* d_ws, size_t ws_size,
                              hipStream_t stream) {
  (void)in_sizes; (void)n_in; (void)out_size; (void)ws_size;

  const float* Q     = (const float*)d_in[0];   // (B, M, QS)
  const float* K     = (const float*)d_in[1];   // (B, N, KS)
  const float* V     = (const float*)d_in[2];   // (B, N, VS)
  const int*   vlen  = (const int*)  d_in[3];   // (B,)
  const float* Wq    = (const float*)d_in[4];   // (QS, H)
  const float* Wk    = (const float*)d_in[5];   // (KS, H)
  const float* wv    = (const float*)d_in[6];   // (H,)
  float*       out   = (float*)d_out;           // (B, M, VS)

  float* Qt   = (float*)d_ws;                     // B*M*H   = 262144 floats
  float* Kt   = Qt + (size_t)BB * MM * HH;        // B*N*H   = 262144 floats
  float* attn = Kt + (size_t)BB * NN * HH;        // B*M*N   = 2097152 floats

  // Qt = Q(4096x256) @ Wq(256x64)
  gemm_wmma_f32_16x16x4<<<dim3((BB * MM) / 16, HH / 16, 1), 32, 0, stream>>>(
      Q, Wq, Qt, QSS, HH, HH, QSS, 0L, 0L, 0L);

  // Kt = K(4096x256) @ Wk(256x64)
  gemm_wmma_f32_16x16x4<<<dim3((BB * NN) / 16, HH / 16, 1), 32, 0, stream>>>(
      K, Wk, Kt, QSS, HH, HH, QSS, 0L, 0L, 0L);

  // attn = softmax(mask(sum_h tanh(Qt+Kt)*wv))
  additive_scores_softmax<<<dim3((BB * MM) / 8), 256, 0, stream>>>(
      Qt, Kt, wv, vlen, attn);

  // out[b] = attn[b](512x512) @ V[b](512x256)
  gemm_wmma_f32_16x16x4<<<dim3(MM / 16, VSS / 16, BB), 32, 0, stream>>>(
      attn, V, out, NN, VSS, VSS, NN,
      (long)MM * NN, (long)NN * VSS, (long)MM * VSS);
}